// GptOssTopKRouter_20469814132796
// MI455X (gfx1250) — compile-verified
//
#include <hip/hip_runtime.h>
#include <math.h>

// ---------------------------------------------------------------------------
// GptOss Top-K router for MI455X (gfx1250, wave32, WMMA f32, async LDS copy)
//   logits = x @ W^T + b ; top-4 ; softmax(top4) ; expert histogram
// ---------------------------------------------------------------------------

#define T_TOKENS    8192
#define HIDDEN      2880
#define NUM_EXPERTS 128
#define TOPK        4

#define TOK_PER_WG  32          // 2 token tiles of 16 per workgroup
#define KC          32          // K chunk staged in LDS (8 WMMA k-steps of 4)
#define NCHUNK      (HIDDEN / KC)   // 90
#define PAD_ROW     36          // 16B-aligned rows, bank-conflict-free stride
#define LOGITS_PAD  129         // bank = (t + e) % 64 -> conflict-free scan

typedef float v2f __attribute__((ext_vector_type(2)));
typedef float v8f __attribute__((ext_vector_type(8)));
typedef int   v4i __attribute__((ext_vector_type(4)));

#if defined(__HIP_DEVICE_COMPILE__)
#if !__has_builtin(__builtin_amdgcn_wmma_f32_16x16x4_f32)
#error "missing __builtin_amdgcn_wmma_f32_16x16x4_f32 on this toolchain"
#endif
#if __has_builtin(__builtin_amdgcn_global_load_async_to_lds_b128)
#define USE_ASYNC_COPY 1
#else
#define USE_ASYNC_COPY 0
#endif
#else
#define USE_ASYNC_COPY 0
#endif

__device__ __forceinline__ void wait_async_zero() {
#if defined(__HIP_DEVICE_COMPILE__)
#if __has_builtin(__builtin_amdgcn_s_wait_asynccnt)
    __builtin_amdgcn_s_wait_asynccnt(0);
#else
    asm volatile("s_wait_asynccnt 0x0" ::: "memory");
#endif
#endif
}

__global__ void __launch_bounds__(256)
router_zero_hist_kernel(float* __restrict__ hist) {
    if (threadIdx.x < NUM_EXPERTS) hist[threadIdx.x] = 0.0f;
}

__global__ void __launch_bounds__(256)
router_kernel(const float* __restrict__ x,        // (T, H)
              const float* __restrict__ weight,   // (E, H)
              const float* __restrict__ bias,     // (E,)
              float* __restrict__ out_scores,     // (T, 4)
              float* __restrict__ out_indices,    // (T, 4) stored as float
              float* __restrict__ out_hist)       // (E,)
{
    __shared__ float xs[2][TOK_PER_WG][PAD_ROW];   // x tiles: 32 tok x 32 k
    __shared__ float ws[2][NUM_EXPERTS][PAD_ROW];  // W tiles: 128 exp x 32 k
    __shared__ float lg[TOK_PER_WG][LOGITS_PAD];   // logits: 32 tok x 128 exp
    __shared__ int   hist[NUM_EXPERTS];

    const int tid     = threadIdx.x;
    const int wave    = tid >> 5;                  // 0..7 -> expert tile
    const int lane    = tid & 31;
    const int lane_lo = lane & 15;
    const int hi      = lane >> 4;                 // K-half selector (f32 WMMA layout)
    const int tok0    = blockIdx.x * TOK_PER_WG;
    const int e0      = wave * 16;

    // staging geometry: thread -> one float4 of x tile, four float4 of W tile
    const int srow = tid >> 3;                     // 0..31
    const int sc4  = (tid & 7) << 2;               // 0,4,...,28

    if (tid < NUM_EXPERTS) hist[tid] = 0;

    v8f acc0 = {};   // tokens tok0+0..15  x experts e0..e0+15
    v8f acc1 = {};   // tokens tok0+16..31 x experts e0..e0+15

#if USE_ASYNC_COPY
    // ---- async DMA staging (GLOBAL_LOAD_ASYNC_TO_LDS_B128, ASYNCcnt) ------
    // builtin signature (from toolchain): (v4i AS1*, v4i AS3*, imm offset, imm cpol)
    typedef v4i __attribute__((address_space(1)))* gvp;
    typedef v4i __attribute__((address_space(3)))* lvp;

    auto issue_chunk = [&](int buf, int kc) {
        __builtin_amdgcn_global_load_async_to_lds_b128(
            (gvp)(x + (size_t)(tok0 + srow) * HIDDEN + kc + sc4),
            (lvp)&xs[buf][srow][sc4], 0, 0);
        #pragma unroll
        for (int i = 0; i < 4; ++i) {
            const int q = tid + (i << 8);
            const int r = q >> 3, c = (q & 7) << 2;
            __builtin_amdgcn_global_load_async_to_lds_b128(
                (gvp)(weight + (size_t)r * HIDDEN + kc + c),
                (lvp)&ws[buf][r][c], 0, 0);
        }
    };

    issue_chunk(0, 0);
    wait_async_zero();
    __syncthreads();

    for (int c = 0; c < NCHUNK; ++c) {
        const int cur = c & 1;
        if (c + 1 < NCHUNK) issue_chunk(cur ^ 1, (c + 1) * KC);  // overlap w/ WMMA
        #pragma unroll
        for (int kk = 0; kk < KC; kk += 4) {
            const int col = kk + (hi << 1);
            const v2f b  = *(const v2f*)&ws[cur][e0 + lane_lo][col];
            const v2f a0 = *(const v2f*)&xs[cur][lane_lo][col];
            const v2f a1 = *(const v2f*)&xs[cur][16 + lane_lo][col];
            acc0 = __builtin_amdgcn_wmma_f32_16x16x4_f32(
                       false, a0, false, b, (short)0, acc0, false, false);
            acc1 = __builtin_amdgcn_wmma_f32_16x16x4_f32(
                       false, a1, false, b, (short)0, acc1, false, false);
        }
        wait_async_zero();
        __syncthreads();
    }
#else
    // ---- fallback: batched register staging, software-pipelined -----------
    float4 vx, vw0, vw1, vw2, vw3;
    auto load_chunk = [&](int kc) {
        vx  = *(const float4*)(x + (size_t)(tok0 + srow) * HIDDEN + kc + sc4);
        {
            const int q = tid;            const int r = q >> 3, cc = (q & 7) << 2;
            vw0 = *(const float4*)(weight + (size_t)r * HIDDEN + kc + cc);
        }
        {
            const int q = tid + 256;      const int r = q >> 3, cc = (q & 7) << 2;
            vw1 = *(const float4*)(weight + (size_t)r * HIDDEN + kc + cc);
        }
        {
            const int q = tid + 512;      const int r = q >> 3, cc = (q & 7) << 2;
            vw2 = *(const float4*)(weight + (size_t)r * HIDDEN + kc + cc);
        }
        {
            const int q = tid + 768;      const int r = q >> 3, cc = (q & 7) << 2;
            vw3 = *(const float4*)(weight + (size_t)r * HIDDEN + kc + cc);
        }
    };
    auto store_chunk = [&](int buf) {
        *(float4*)&xs[buf][srow][sc4] = vx;   // PAD_ROW=36 keeps 16B alignment
        { const int q = tid;       *(float4*)&ws[buf][q >> 3][(q & 7) << 2] = vw0; }
        { const int q = tid + 256; *(float4*)&ws[buf][q >> 3][(q & 7) << 2] = vw1; }
        { const int q = tid + 512; *(float4*)&ws[buf][q >> 3][(q & 7) << 2] = vw2; }
        { const int q = tid + 768; *(float4*)&ws[buf][q >> 3][(q & 7) << 2] = vw3; }
    };

    load_chunk(0);
    store_chunk(0);
    __syncthreads();

    for (int c = 0; c < NCHUNK; ++c) {
        const int cur = c & 1;
        const bool more = (c + 1 < NCHUNK);
        if (more) load_chunk((c + 1) * KC);   // loads in flight during WMMA
        #pragma unroll
        for (int kk = 0; kk < KC; kk += 4) {
            const int col = kk + (hi << 1);
            const v2f b  = *(const v2f*)&ws[cur][e0 + lane_lo][col];
            const v2f a0 = *(const v2f*)&xs[cur][lane_lo][col];
            const v2f a1 = *(const v2f*)&xs[cur][16 + lane_lo][col];
            acc0 = __builtin_amdgcn_wmma_f32_16x16x4_f32(
                       false, a0, false, b, (short)0, acc0, false, false);
            acc1 = __builtin_amdgcn_wmma_f32_16x16x4_f32(
                       false, a1, false, b, (short)0, acc1, false, false);
        }
        if (more) store_chunk(cur ^ 1);
        __syncthreads();
    }
#endif

    // ---- spill logits (+bias) to LDS.
    // C/D layout: lane(0..15)->N, VGPR r -> M=r (lanes 0-15) / M=r+8 (lanes 16-31)
    const float bN = bias[e0 + lane_lo];
    #pragma unroll
    for (int r = 0; r < 8; ++r) {
        const int m = r + (hi << 3);
        lg[m][e0 + lane_lo]      = acc0[r] + bN;
        lg[16 + m][e0 + lane_lo] = acc1[r] + bN;
    }
    __syncthreads();

    // ---- per-token top-4 + softmax + LDS histogram (32 token threads)
    if (tid < TOK_PER_WG) {
        float v[TOPK] = {-INFINITY, -INFINITY, -INFINITY, -INFINITY};
        int   ix[TOPK] = {0, 0, 0, 0};
        for (int e = 0; e < NUM_EXPERTS; ++e) {
            const float lv = lg[tid][e];
            if (lv > v[TOPK - 1]) {
                v[TOPK - 1] = lv; ix[TOPK - 1] = e;
                #pragma unroll
                for (int j = TOPK - 1; j > 0; --j) {
                    if (v[j] > v[j - 1]) {          // strict > : lowest index wins ties
                        const float tv = v[j]; v[j] = v[j - 1]; v[j - 1] = tv;
                        const int   ti = ix[j]; ix[j] = ix[j - 1]; ix[j - 1] = ti;
                    }
                }
            }
        }
        float ex[TOPK], s = 0.0f;
        #pragma unroll
        for (int k = 0; k < TOPK; ++k) { ex[k] = __expf(v[k] - v[0]); s += ex[k]; }
        const float inv = 1.0f / s;
        const int token = tok0 + tid;
        #pragma unroll
        for (int k = 0; k < TOPK; ++k) {
            out_scores[(size_t)token * TOPK + k]  = ex[k] * inv;
            out_indices[(size_t)token * TOPK + k] = (float)ix[k];
            atomicAdd(&hist[ix[k]], 1);
        }
    }
    __syncthreads();

    // ---- one global f32 atomic per expert per workgroup
    if (tid < NUM_EXPERTS) {
        const int c = hist[tid];
        if (c) atomicAdd(&out_hist[tid], (float)c);
    }
}

extern "C" void kernel_launch(void* const* d_in, const int* in_sizes, int n_in,
                              void* d_out, int out_size, void* d_ws, size_t ws_size,
                              hipStream_t stream) {
    (void)in_sizes; (void)n_in; (void)d_ws; (void)ws_size; (void)out_size;

    const float* x      = (const float*)d_in[0];   // (8192, 2880) f32
    const float* weight = (const float*)d_in[1];   // (128, 2880)  f32
    const float* bias   = (const float*)d_in[2];   // (128,)       f32

    float* out     = (float*)d_out;
    float* scores  = out;                               // 8192*4
    float* indices = out + (size_t)T_TOKENS * TOPK;     // 8192*4
    float* histo   = out + (size_t)2 * T_TOKENS * TOPK; // 128

    // Deterministic across graph replays: zero the histogram every call.
    router_zero_hist_kernel<<<1, 256, 0, stream>>>(histo);

    router_kernel<<<T_TOKENS / TOK_PER_WG, 256, 0, stream>>>(
        x, weight, bias, scores, indices, histo);
}